// SnnLocalConvNet_74526272520630
// MI455X (gfx1250) — compile-verified
//
#include <hip/hip_runtime.h>

typedef __attribute__((ext_vector_type(2))) float v2f;
typedef __attribute__((ext_vector_type(8))) float v8f;

#define BB   32
#define CIN  64
#define COUT 64
#define HO   30
#define NPOS 900          // 30*30
#define KT   9            // 3x3 taps
#define TOT  (BB*COUT*NPOS)   // 1,843,200 per output tensor

// workspace layout (float offsets)
#define WS_Y     0
#define WS_PART  (TOT)                   // 900*64 sums
#define WS_PART2 (TOT + NPOS*COUT)       // 900*64 sumsq
#define WS_BN    (TOT + 2*NPOS*COUT)     // 64 scale + 64 shift

#define AP 74   // LDS pitch for 32x72 A tile (even -> 8B aligned float2)
#define BP 74   // LDS pitch for 64x72 B^T tile
#define SP 66   // LDS pitch for 64-wide tiles in kernel 3

// ---------------- Kernel 1: locally-connected conv (WMMA f32) ----------------
// grid = 900 positions, block = 256 (8 waves; wave w -> Mtile w&1, Ntile w>>1)
__global__ __launch_bounds__(256) void k1_conv(
    const float* __restrict__ ff, const float* __restrict__ cw,
    const float* __restrict__ cb, float* __restrict__ ws)
{
  __shared__ float Al[BB * AP];     // patches: A[b][ck]
  __shared__ float Bl[COUT * BP];   // weights^T: B[o][ck]
  __shared__ float redS[256];
  __shared__ float redQ[256];

  const int p = blockIdx.x;
  const int h = p / HO, w = p % HO;
  const int t = threadIdx.x;
  const int lane = t & 31, wave = t >> 5;
  const int mt = wave & 1, nt = wave >> 1;
  const int ln = lane & 15, hi = lane >> 4;
  const int m = mt * 16 + ln;   // batch row of A fragment
  const int n = nt * 16 + ln;   // out-channel col of B fragment

  v8f acc = {};
  for (int cg = 0; cg < 8; ++cg) {          // 8 groups of 8 input channels
    __syncthreads();
    {   // A: 32 batches x 8 channels x 9 taps (one patch per thread)
      int b = t >> 3, cc = t & 7, c = cg * 8 + cc;
      const float* src = ff + ((b * CIN + c) * 32 + h) * 32 + w;
      float* dst = Al + b * AP + cc * KT;
      #pragma unroll
      for (int kh = 0; kh < 3; ++kh)
        #pragma unroll
        for (int kw = 0; kw < 3; ++kw)
          dst[kh * 3 + kw] = src[kh * 32 + kw];
    }
    {   // B: 64 couts x 8 channels x 9 taps (two patches per thread)
      #pragma unroll
      for (int rep = 0; rep < 2; ++rep) {
        int idx = t + rep * 256;
        int o = idx >> 3, cc = idx & 7, c = cg * 8 + cc;
        const float* src = cw + ((size_t)(o * CIN + c) * NPOS + p) * KT;
        float* dst = Bl + o * BP + cc * KT;
        #pragma unroll
        for (int k = 0; k < KT; ++k) dst[k] = src[k];
      }
    }
    __syncthreads();
    #pragma unroll
    for (int kb = 0; kb < 72; kb += 4) {    // 18 WMMAs / chunk, 144 total
      v2f av = *(const v2f*)(Al + m * AP + kb + 2 * hi);
      v2f bv = *(const v2f*)(Bl + n * BP + kb + 2 * hi);
      acc = __builtin_amdgcn_wmma_f32_16x16x4_f32(false, av, false, bv,
                                                  (short)0, acc, false, false);
    }
  }

  // bias, store y, deterministic per-channel partial sums for BN
  const float bias = cb[n * NPOS + p];
  float s = 0.f, q = 0.f;
  #pragma unroll
  for (int r = 0; r < 8; ++r) {
    float v = acc[r] + bias;
    int b = mt * 16 + hi * 8 + r;           // C/D layout: M = 8*hi + r
    ws[WS_Y + (b * COUT + n) * NPOS + p] = v;
    s += v; q += v * v;
  }
  redS[t] = s; redQ[t] = q;
  __syncthreads();
  if (t < COUT) {                            // fixed-order 4-term fold
    int nt2 = t >> 4, l0 = t & 15;
    float S = 0.f, Q = 0.f;
    #pragma unroll
    for (int mt2 = 0; mt2 < 2; ++mt2)
      #pragma unroll
      for (int h2 = 0; h2 < 2; ++h2) {
        int tt = ((nt2 * 2 + mt2) << 5) + h2 * 16 + l0;
        S += redS[tt]; Q += redQ[tt];
      }
    ws[WS_PART  + p * COUT + t] = S;
    ws[WS_PART2 + p * COUT + t] = Q;
  }
}

// ---------------- Kernel 2: finalize BN scale/shift ----------------
__global__ __launch_bounds__(64) void k2_bn(
    const float* __restrict__ gamma, const float* __restrict__ beta,
    float* __restrict__ ws)
{
  const int ch = threadIdx.x;
  float S = 0.f, Q = 0.f;
  for (int p = 0; p < NPOS; ++p) {
    S += ws[WS_PART  + p * COUT + ch];
    Q += ws[WS_PART2 + p * COUT + ch];
  }
  const float inv = 1.0f / (float)(BB * NPOS);
  const float mean = S * inv;
  const float var  = Q * inv - mean * mean;
  const float sc = gamma[ch] * rsqrtf(var + 1e-5f);
  ws[WS_BN + ch]        = sc;
  ws[WS_BN + COUT + ch] = beta[ch] - mean * sc;
}

// ---------------- Kernel 3: recurrence GEMM (WMMA) + adaptive LIF ----------------
__global__ __launch_bounds__(256) void k3_rec_lif(
    const float* __restrict__ fbv,  const float* __restrict__ soma,
    const float* __restrict__ spk,  const float* __restrict__ acur,
    const float* __restrict__ bt,   const float* __restrict__ lrec,
    const float* __restrict__ taum, const float* __restrict__ taud,
    const float* __restrict__ taua, const float* __restrict__ ws,
    float* __restrict__ out)
{
  __shared__ float As[BB * SP];     // spk[b][d]
  __shared__ float Bs[COUT * SP];   // local_rec[c][d]  (== B^T[n][k])
  const int p = blockIdx.x;
  const int t = threadIdx.x;
  const int lane = t & 31, wave = t >> 5;
  const int mt = wave & 1, nt = wave >> 1;
  const int ln = lane & 15, hi = lane >> 4;
  const int m = mt * 16 + ln;
  const int n = nt * 16 + ln;

  #pragma unroll
  for (int i = 0; i < 8; ++i) {     // 32x64 spk tile (strided gather)
    int e = i * 256 + t;
    int b = e >> 6, d = e & 63;
    As[b * SP + d] = spk[(b * COUT + d) * NPOS + p];
  }
  const float* lp = lrec + (size_t)p * COUT * COUT;
  #pragma unroll
  for (int i = 0; i < 16; ++i) {    // 64x64 local_rec tile (contiguous)
    int e = i * 256 + t;
    int c = e >> 6, d = e & 63;
    Bs[c * SP + d] = lp[e];
  }
  __syncthreads();

  v8f acc = {};
  #pragma unroll
  for (int kb = 0; kb < 64; kb += 4) {     // rec[b,c] = sum_d spk[b,d]*L[c,d]
    v2f av = *(const v2f*)(As + m * SP + kb + 2 * hi);
    v2f bv = *(const v2f*)(Bs + n * SP + kb + 2 * hi);
    acc = __builtin_amdgcn_wmma_f32_16x16x4_f32(false, av, false, bv,
                                                (short)0, acc, false, false);
  }

  const float sc = ws[WS_BN + n];
  const float sh = ws[WS_BN + COUT + n];
  const float alpha = __expf(-0.5f / taum[n * NPOS + p]);   // DT = 0.5
  const float rho   = __expf(-0.5f / taud[n * NPOS + p]);
  const float eta   = __expf(-0.5f / taua[n * NPOS + p]);

  #pragma unroll
  for (int r = 0; r < 8; ++r) {
    int b = mt * 16 + hi * 8 + r;
    int idx = (b * COUT + n) * NPOS + p;
    float sv    = As[b * SP + n];
    float convx = ws[WS_Y + idx] * sc + sh + acc[r];
    float bnew  = rho * bt[idx] + (1.f - rho) * sv;
    float thre  = 0.1f + 1.8f * bnew;                 // B_J0 + BETA*b
    float anew  = eta * acur[idx] + fbv[idx];
    float sg    = 1.f / (1.f + __expf(-anew)) - 0.5f; // shifted sigmoid
    float snew  = alpha * soma[idx] + sg + convx - thre * sv;
    float spike = (snew - thre) > 0.f ? 1.f : 0.f;
    out[idx]           = snew;
    out[TOT   + idx]   = spike;
    out[2*TOT + idx]   = anew;
    out[3*TOT + idx]   = bnew;
  }
}

extern "C" void kernel_launch(void* const* d_in, const int* in_sizes, int n_in,
                              void* d_out, int out_size, void* d_ws, size_t ws_size,
                              hipStream_t stream) {
  const float* ff   = (const float*)d_in[0];
  const float* fb   = (const float*)d_in[1];
  const float* soma = (const float*)d_in[2];
  const float* spk  = (const float*)d_in[3];
  const float* acur = (const float*)d_in[4];
  const float* bt   = (const float*)d_in[5];
  const float* cw   = (const float*)d_in[6];
  const float* cb   = (const float*)d_in[7];
  const float* gam  = (const float*)d_in[8];
  const float* bet  = (const float*)d_in[9];
  const float* lrec = (const float*)d_in[10];
  const float* taum = (const float*)d_in[11];
  const float* taud = (const float*)d_in[12];
  const float* taua = (const float*)d_in[13];
  float* out = (float*)d_out;
  float* ws  = (float*)d_ws;

  hipLaunchKernelGGL(k1_conv,    dim3(NPOS), dim3(256), 0, stream, ff, cw, cb, ws);
  hipLaunchKernelGGL(k2_bn,      dim3(1),    dim3(64),  0, stream, gam, bet, ws);
  hipLaunchKernelGGL(k3_rec_lif, dim3(NPOS), dim3(256), 0, stream,
                     fb, soma, spk, acur, bt, lrec, taum, taud, taua, ws, out);
}